// MLKVLinearAttnDecodeStep_10969346474366
// MI455X (gfx1250) — compile-verified
//
#include <hip/hip_runtime.h>
#include <math.h>

// MLKV linear-attention decode step for gfx1250 (MI455X).
// Memory-bound (~141 MB fp32 traffic -> ~6us @ 23.3 TB/s). GEMVs use
// coalesced b128 streaming + global_prefetch; the per-head rec einsums use
// V_WMMA_F32_16X16X4_F32 with B fragments served by unconditional ds_load_b64.

#define H_DIM     2048
#define NUMV      32
#define NUMK      16
#define DKD       128
#define DVD       128
#define KEY_DIM   2048
#define VALUE_DIM 4096
#define CONV_DIM  8192

typedef __attribute__((ext_vector_type(2))) float v2f;
typedef __attribute__((ext_vector_type(8))) float v8f;

// workspace layout (float offsets)
#define WS_MIXED 0         // 8192
#define WS_Z     8192      // 4096
#define WS_B     12288     // 32
#define WS_A     12320     // 32
#define WS_CONV  12352     // 8192
#define WS_O     20544     // 4096
#define WS_BETA  24640     // 32
#define WS_GEXP  24672     // 32
#define WS_KQ    24704     // 16
#define WS_QN    24720     // 2048
#define WS_KN    26768     // 2048
#define WS_DELTA 28816     // 4096

// ---------------- Kernel 1: fused GEMV for W_qkv / W_z / W_b / W_a ----------
// wave-per-row, h staged in LDS, b128 weight streaming (bandwidth-bound).
__global__ void k1_gemv_qkvzba(const float* __restrict__ h,
                               const float* __restrict__ Wqkv,
                               const float* __restrict__ Wz,
                               const float* __restrict__ Wb,
                               const float* __restrict__ Wa,
                               float* __restrict__ ws) {
  __shared__ float sh[H_DIM];
  const int tid = threadIdx.x;
  const float4* h4 = (const float4*)h;
  float4* sh4 = (float4*)sh;
  for (int i = tid; i < H_DIM / 4; i += 256) sh4[i] = h4[i];
  __syncthreads();
  const int wave = tid >> 5, lane = tid & 31;
  const int r = blockIdx.x * 8 + wave;   // 1544*8 = 12352 rows exactly
  const float* Wrow;
  float* dst;
  if (r < CONV_DIM) {
    Wrow = Wqkv + (size_t)r * H_DIM;                 dst = ws + WS_MIXED + r;
  } else if (r < CONV_DIM + VALUE_DIM) {
    int rr = r - CONV_DIM;
    Wrow = Wz + (size_t)rr * H_DIM;                  dst = ws + WS_Z + rr;
  } else if (r < CONV_DIM + VALUE_DIM + NUMV) {
    int rr = r - CONV_DIM - VALUE_DIM;
    Wrow = Wb + (size_t)rr * H_DIM;                  dst = ws + WS_B + rr;
  } else {
    int rr = r - CONV_DIM - VALUE_DIM - NUMV;
    Wrow = Wa + (size_t)rr * H_DIM;                  dst = ws + WS_A + rr;
  }
  const float4* W4 = (const float4*)Wrow;
  float s = 0.f;
  for (int j = lane; j < H_DIM / 4; j += 32) {
    __builtin_prefetch(W4 + j + 96, 0, 0);           // global_prefetch ~3 iters ahead
    float4 w = W4[j]; float4 x = sh4[j];
    s += w.x * x.x + w.y * x.y + w.z * x.z + w.w * x.w;
  }
#pragma unroll
  for (int o = 16; o > 0; o >>= 1) s += __shfl_xor(s, o, 32);
  if (lane == 0) *dst = s;
}

// ---------------- Kernel 2: conv-state shift + depthwise conv + SiLU --------
__global__ void k2_conv(const float* __restrict__ conv_state,
                        const float* __restrict__ conv_w,
                        float* __restrict__ ws,
                        float* __restrict__ out_conv) {
  const int c = blockIdx.x * 256 + threadIdx.x;       // 8192 channels
  float4 cs = ((const float4*)conv_state)[c];
  float4 cw = ((const float4*)conv_w)[c];
  float mx = ws[WS_MIXED + c];
  float4 ncs; ncs.x = cs.y; ncs.y = cs.z; ncs.z = cs.w; ncs.w = mx;
  float acc = ncs.x * cw.x + ncs.y * cw.y + ncs.z * cw.z + ncs.w * cw.w;
  ws[WS_CONV + c] = acc / (1.f + expf(-acc));         // silu
  ((float4*)out_conv)[c] = ncs;
}

// ---------------- Kernel 3: q/k l2norm per k-head + gate scalars ------------
__global__ void k3_norm_scalars(const float* __restrict__ dt_bias,
                                const float* __restrict__ A_log,
                                float* __restrict__ ws) {
  const int tid = threadIdx.x;
  if (blockIdx.x == 16) {                              // scalar block
    if (tid < NUMV) {
      float bb = ws[WS_B + tid];
      ws[WS_BETA + tid] = 1.f / (1.f + expf(-bb));
      float x = ws[WS_A + tid] + dt_bias[tid];
      float sp = (x > 20.f) ? x : log1pf(expf(x));     // softplus
      ws[WS_GEXP + tid] = expf(-expf(A_log[tid]) * sp);
    }
    return;
  }
  const int kh = blockIdx.x;                           // 0..15, 128 threads
  float qraw = ws[WS_CONV + kh * DKD + tid];
  float kraw = ws[WS_CONV + KEY_DIM + kh * DKD + tid];
  float q2 = qraw * qraw, k2 = kraw * kraw, qk = qraw * kraw;
  __shared__ float red[3][4];
  const int wave = tid >> 5, lane = tid & 31;
#pragma unroll
  for (int o = 16; o > 0; o >>= 1) {
    q2 += __shfl_xor(q2, o, 32);
    k2 += __shfl_xor(k2, o, 32);
    qk += __shfl_xor(qk, o, 32);
  }
  if (lane == 0) { red[0][wave] = q2; red[1][wave] = k2; red[2][wave] = qk; }
  __syncthreads();
  float sq  = red[0][0] + red[0][1] + red[0][2] + red[0][3];
  float sk  = red[1][0] + red[1][1] + red[1][2] + red[1][3];
  float sqk = red[2][0] + red[2][1] + red[2][2] + red[2][3];
  float rq = rsqrtf(sq + 1e-6f) * 0.08838834764831845f;  // * 1/sqrt(DK)
  float rk = rsqrtf(sk + 1e-6f);
  ws[WS_QN + kh * DKD + tid] = qraw * rq;
  ws[WS_KN + kh * DKD + tid] = kraw * rk;
  if (tid == 0) ws[WS_KQ + kh] = sqk * rq * rk;          // dot(qn,kn)
}

// ---------------- Kernel 4: per-head state einsums via WMMA f32 16x16x4 -----
// D(16x16) = A(16x4) * B(4x16):
//   A[m][kk] = rec[h][k0+kk][d0+m]   (M = d over lanes -> coalesced)
//   B[kk][0] = kn[k0+kk], B[kk][1] = qn[k0+kk], columns 2..15 zero
// B is staged in a 16x128 LDS array (rows >=2 zero) so every lane issues one
// unconditional ds_load_b64 per WMMA - no exec-mask churn in the hot loop.
// -> column 0 of D accumulates rec.k, column 1 accumulates rec.q.
// Then: delta = (v - gexp*rec.k)*beta ; core = gexp*rec.q + (k.q)*delta.
__global__ void k4_head_update(const float* __restrict__ rec,
                               const float* __restrict__ normw,
                               float* __restrict__ ws) {
  const int hh = blockIdx.x;          // 32 heads
  const int kh = hh >> 1;             // VPK = 2
  const int tid = threadIdx.x;
  const int wave = tid >> 5, lane = tid & 31;
  const int m = lane & 15, half = lane >> 4;
  const int d0 = wave * 16;           // 8 waves cover d = 0..127
  __shared__ float sB[16][DKD];       // B source: row0=kn, row1=qn, rest 0
  __shared__ float sKV[8][16];
  __shared__ float sQC[8][16];
  __shared__ float sRed[8];
  for (int i = tid; i < 16 * DKD; i += 256) ((float*)sB)[i] = 0.f;
  __syncthreads();
  if (tid < DKD) {
    sB[0][tid] = ws[WS_KN + kh * DKD + tid];
    sB[1][tid] = ws[WS_QN + kh * DKD + tid];
  }
  __syncthreads();
  const float* recH = rec + (size_t)hh * (DKD * DVD);
  v8f acc = {0.f, 0.f, 0.f, 0.f, 0.f, 0.f, 0.f, 0.f};
#pragma unroll 4
  for (int k0 = 0; k0 < DKD; k0 += 4) {
    const int kb = k0 + half * 2;     // lanes 0-15: K=0,1 ; lanes 16-31: K=2,3
    v2f a, b;
    a.x = recH[(size_t)kb * DVD + d0 + m];
    a.y = recH[(size_t)(kb + 1) * DVD + d0 + m];
    b = *(const v2f*)&sB[m][kb];      // single ds_load_b64, all lanes active
    acc = __builtin_amdgcn_wmma_f32_16x16x4_f32(false, a, false, b,
                                                (short)0, acc, false, false);
  }
  // C/D layout: column n lives in lane n (M=0..7 in vgpr 0..7) and lane n+16 (M=8..15)
  if (m < 2) {
    float* dstp = m ? &sQC[wave][0] : &sKV[wave][0];
#pragma unroll
    for (int r = 0; r < 8; ++r) dstp[half * 8 + r] = acc[r];
  }
  __syncthreads();
  float core = 0.f;
  const int d = tid;
  if (tid < DVD) {
    float ge = ws[WS_GEXP + hh];
    float be = ws[WS_BETA + hh];
    float kq = ws[WS_KQ + kh];
    float kv = ge * sKV[d >> 4][d & 15];          // (rec*exp(g)) . k
    float qc = ge * sQC[d >> 4][d & 15];          // (rec*exp(g)) . q
    float vv = ws[WS_CONV + 2 * KEY_DIM + hh * DVD + d];
    float dl = (vv - kv) * be;
    core = qc + kq * dl;                          // rec_new . q without re-reading rec
    ws[WS_DELTA + hh * DVD + d] = dl;
  }
  float c2 = (tid < DVD) ? core * core : 0.f;
#pragma unroll
  for (int o = 16; o > 0; o >>= 1) c2 += __shfl_xor(c2, o, 32);
  if (lane == 0) sRed[wave] = c2;
  __syncthreads();
  if (tid < DVD) {
    float mean = (sRed[0] + sRed[1] + sRed[2] + sRed[3]) * (1.f / 128.f);
    float rms = rsqrtf(mean + 1e-6f);
    float xn = core * rms * normw[d];
    float zf = ws[WS_Z + hh * DVD + d];
    float sil = zf / (1.f + expf(-zf));
    ws[WS_O + hh * DVD + d] = xn * sil;
  }
}

// ---------------- Kernel 5: rec_new = rec*exp(g) + k (outer) delta ----------
__global__ void k5_rec_write(const float* __restrict__ rec_in,
                             const float* __restrict__ ws,
                             float* __restrict__ out_rec) {
  const int i4 = blockIdx.x * 256 + threadIdx.x;   // 131072 float4s
  const int base = i4 << 2;
  const int hh = base >> 14;
  const int k  = (base >> 7) & 127;
  const int d  = base & 127;
  float4 r = ((const float4*)rec_in)[i4];
  float ge = ws[WS_GEXP + hh];
  float kn = ws[WS_KN + (hh >> 1) * DKD + k];
  const float4 dl = *(const float4*)(ws + WS_DELTA + hh * DVD + d);
  float4 o;
  o.x = r.x * ge + kn * dl.x;
  o.y = r.y * ge + kn * dl.y;
  o.z = r.z * ge + kn * dl.z;
  o.w = r.w * ge + kn * dl.w;
  ((float4*)out_rec)[i4] = o;
}

// ---------------- Kernel 6: hidden_out = W_out @ o --------------------------
__global__ void k6_out_gemv(const float* __restrict__ Wout,
                            const float* __restrict__ ws,
                            float* __restrict__ out_hidden) {
  __shared__ float so[VALUE_DIM];
  const int tid = threadIdx.x;
  const float4* o4 = (const float4*)(ws + WS_O);
  float4* so4 = (float4*)so;
  for (int i = tid; i < VALUE_DIM / 4; i += 256) so4[i] = o4[i];
  __syncthreads();
  const int wave = tid >> 5, lane = tid & 31;
  const int r = blockIdx.x * 8 + wave;             // 2048 rows
  const float4* W4 = (const float4*)(Wout + (size_t)r * VALUE_DIM);
  float s = 0.f;
  for (int j = lane; j < VALUE_DIM / 4; j += 32) {
    __builtin_prefetch(W4 + j + 96, 0, 0);         // global_prefetch ~3 iters ahead
    float4 w = W4[j]; float4 x = so4[j];
    s += w.x * x.x + w.y * x.y + w.z * x.z + w.w * x.w;
  }
#pragma unroll
  for (int o = 16; o > 0; o >>= 1) s += __shfl_xor(s, o, 32);
  if (lane == 0) out_hidden[r] = s;
}

extern "C" void kernel_launch(void* const* d_in, const int* in_sizes, int n_in,
                              void* d_out, int out_size, void* d_ws, size_t ws_size,
                              hipStream_t stream) {
  (void)in_sizes; (void)n_in; (void)out_size; (void)ws_size;
  const float* h    = (const float*)d_in[0];
  const float* cst  = (const float*)d_in[1];
  const float* rec  = (const float*)d_in[2];
  const float* Wqkv = (const float*)d_in[3];
  const float* Wz   = (const float*)d_in[4];
  const float* Wb   = (const float*)d_in[5];
  const float* Wa   = (const float*)d_in[6];
  const float* Wout = (const float*)d_in[7];
  const float* cw   = (const float*)d_in[8];
  const float* dtb  = (const float*)d_in[9];
  const float* Alog = (const float*)d_in[10];
  const float* nw   = (const float*)d_in[11];
  float* out = (float*)d_out;
  float* ws  = (float*)d_ws;
  float* out_hidden = out;                   // 2048
  float* out_conv   = out + 2048;            // 32768
  float* out_rec    = out + 2048 + 32768;    // 524288

  k1_gemv_qkvzba<<<1544, 256, 0, stream>>>(h, Wqkv, Wz, Wb, Wa, ws);
  k2_conv<<<CONV_DIM / 256, 256, 0, stream>>>(cst, cw, ws, out_conv);
  k3_norm_scalars<<<17, 128, 0, stream>>>(dtb, Alog, ws);
  k4_head_update<<<NUMV, 256, 0, stream>>>(rec, nw, ws);
  k5_rec_write<<<512, 256, 0, stream>>>(rec, ws, out_rec);
  k6_out_gemv<<<256, 256, 0, stream>>>(Wout, ws, out_hidden);
}